// ElasticInteractionEnergyLoss_51204600103262
// MI455X (gfx1250) — compile-verified
//
#include <hip/hip_runtime.h>
#include <hip/hip_bf16.h>
#include <math.h>

typedef __attribute__((ext_vector_type(2))) float v2f;
typedef __attribute__((ext_vector_type(8))) float v8f;
typedef __attribute__((ext_vector_type(4))) unsigned int v4u;
typedef __attribute__((ext_vector_type(8))) int v8i;
typedef __attribute__((ext_vector_type(4))) int v4i;

constexpr int   FDIM  = 224;
constexpr int   FF    = FDIM * FDIM;
constexpr int   B     = 64;
constexpr int   L     = 8;
constexpr int   P     = 72;
constexpr int   SEG   = P - 1;              // 71
constexpr int   T_MAX = 2 * (FDIM - 1);     // 446
constexpr float EPS   = 1e-8f;
constexpr float ALPHA = 1.0f;
constexpr int   TILES = FDIM / 16;          // 14 tiles per dim
constexpr int   TPB   = TILES * TILES;      // 196 tiles per batch
constexpr int   GTHREADS = TILES * 32;      // 448: one wave per m-tile

// workspace layout (in floats)
constexpr size_t OFF_GT   = 0;                            // diff computed in-place here
constexpr size_t OFF_PRED = OFF_GT   + (size_t)B * FF;
constexpr size_t OFF_A    = OFF_PRED + (size_t)B * FF;    // C*diff
constexpr size_t OFF_BM   = OFF_A    + (size_t)B * FF;    // S*diff
constexpr size_t OFF_C    = OFF_BM   + (size_t)B * FF;    // cos DFT matrix
constexpr size_t OFF_S    = OFF_C    + (size_t)FF;        // -sin DFT matrix
constexpr size_t OFF_PART = OFF_S    + (size_t)FF;        // per-wave partial sums

// ---------------------------------------------------------------------------
// TDM: DMA a 2-D column tile (rows x cols f32, row stride in elements) from
// global memory into LDS, laid out linearly [row][col]. D# layout per CDNA5
// ISA ch.8: group0 = count/lds_addr/global_addr/type, group1 = dims/strides,
// groups 2-3 zero (2-D tensor). 6-arg clang-23 builtin form (extra v8i group).
// Issued once per wave; EXEC is ignored.
// ---------------------------------------------------------------------------
__device__ __forceinline__ void tdm_load_tile_f32(const float* gsrc,
                                                  unsigned lds_off,
                                                  int rows, int cols,
                                                  int row_stride_elems) {
  unsigned long long ga = (unsigned long long)(uintptr_t)gsrc;
  v4u g0;
  g0.x = 1u;                                              // count=1, user descriptor
  g0.y = lds_off;                                         // lds_addr (bytes)
  g0.z = (unsigned)(ga & 0xFFFFFFFFull);                  // global_addr[31:0]
  g0.w = (unsigned)((ga >> 32) & 0x01FFFFFFull)           // global_addr[56:32]
         | (2u << 30);                                    // type=2 ("image")
  v8i g1;
  g1[0] = (int)(2u << 16);                 // wg_mask=0 | data_size=2 (4 bytes)
  g1[1] = (int)((unsigned)cols << 16);     // tensor_dim0[15:0]  (bits 63:48)
  g1[2] = (int)((unsigned)rows << 16);     // tensor_dim1[15:0]  (bits 95:80)
  g1[3] = (int)((unsigned)cols << 16);     // tile_dim0          (bits 127:112)
  g1[4] = rows;                            // tile_dim1          (bits 143:128)
  g1[5] = row_stride_elems;                // tensor_dim0_stride[31:0]
  g1[6] = 0;
  g1[7] = 0;
  v4i g2 = {0, 0, 0, 0};
  v4i g3 = {0, 0, 0, 0};
  v8i g4 = {0, 0, 0, 0, 0, 0, 0, 0};       // extended group (unused, 2-D tile)
  __builtin_amdgcn_tensor_load_to_lds(g0, g1, g2, g3, g4, 0);
}

// ---------------------------------------------------------------------------
// Build the 224x224 complex DFT matrix F = C + iS, F[k,n] = exp(-2*pi*i*k*n/F)
// ---------------------------------------------------------------------------
__global__ void build_dft(float* __restrict__ ws) {
  int idx = blockIdx.x * blockDim.x + threadIdx.x;
  if (idx >= FF) return;
  int k = idx / FDIM, n = idx % FDIM;
  int ph = (k * n) % FDIM;   // exploit periodicity for full f32 accuracy
  float theta = 6.28318530717958647692f * (float)ph / (float)FDIM;
  ws[OFF_C + idx] = cosf(theta);
  ws[OFF_S + idx] = -sinf(theta);
}

// ---------------------------------------------------------------------------
// Rasterize one (batch, line) per block; blockIdx.y selects gt(0)/pred(1).
// Segment params staged in LDS; scatter of 1.0 is idempotent -> race-free.
// ---------------------------------------------------------------------------
__global__ void rasterize(const float* __restrict__ pred,
                          const float* __restrict__ gt,
                          const unsigned char* __restrict__ vmask,
                          float* __restrict__ ws) {
  int b = blockIdx.x / L, l = blockIdx.x % L;
  bool isPred = (blockIdx.y == 1);
  const float* kps = (isPred ? pred : gt) + (size_t)(b * L + l) * P * 2;
  const unsigned char* vm = vmask + (size_t)(b * L + l) * P;
  float* field = ws + (isPred ? OFF_PRED : OFF_GT) + (size_t)b * FF;

  __shared__ float sx1[SEG], sy1[SEG], sx2[SEG], sy2[SEG];
  __shared__ int   sn[SEG], sv[SEG];

  for (int s = threadIdx.x; s < SEG; s += blockDim.x) {
    float p1x = kps[s * 2 + 0], p1y = kps[s * 2 + 1];
    float p2x = kps[s * 2 + 2], p2y = kps[s * 2 + 3];
    bool valid = (vm[s] != 0) && (vm[s + 1] != 0) &&
                 p1x >= 0.f && p1x <= 1.f && p1y >= 0.f && p1y <= 1.f &&
                 p2x >= 0.f && p2x <= 1.f && p2y >= 0.f && p2y <= 1.f;
    int x1 = (int)floorf(p1x * (FDIM - 1));
    int y1 = (int)floorf(p1y * (FDIM - 1));
    int x2 = (int)floorf(p2x * (FDIM - 1));
    int y2 = (int)floorf(p2y * (FDIM - 1));
    int ax = abs(x2 - x1), ay = abs(y2 - y1);
    int n = 2 * (ax > ay ? ax : ay);
    if (n < 2) n = 2;
    sx1[s] = (float)x1; sy1[s] = (float)y1;
    sx2[s] = (float)x2; sy2[s] = (float)y2;
    sn[s] = n; sv[s] = valid ? 1 : 0;
  }
  __syncthreads();

  int total = SEG * T_MAX;
  for (int idx = threadIdx.x; idx < total; idx += blockDim.x) {
    int s = idx / T_MAX, k = idx % T_MAX;
    if (!sv[s] || k >= sn[s]) continue;
    int den = sn[s] - 1; if (den < 1) den = 1;
    float t = (float)k / (float)den;
    int lx = (int)rintf(sx1[s] * (1.f - t) + sx2[s] * t);  // RNE matches jnp.round
    int ly = (int)rintf(sy1[s] * (1.f - t) + sy2[s] * t);
    #pragma unroll
    for (int dy = -1; dy <= 1; dy++)
      #pragma unroll
      for (int dx = -1; dx <= 1; dx++) {
        int x = lx + dx, y = ly + dy;
        if (x >= 0 && x < FDIM && y >= 0 && y < FDIM) field[y * FDIM + x] = 1.0f;
      }
  }
}

// diff = gt - ALPHA*pred, in-place into the gt buffer
__global__ void diff_kernel(float* __restrict__ ws) {
  size_t idx = (size_t)blockIdx.x * blockDim.x + threadIdx.x;
  if (idx < (size_t)B * FF) ws[OFF_GT + idx] -= ALPHA * ws[OFF_PRED + idx];
}

// ---------------------------------------------------------------------------
// Stage 1: A = C*diff, Bm = S*diff. One block per (batch, n-tile); wave w
// computes m-tile w. The 224x16 diff column tile is DMA'd into LDS by the
// Tensor Data Mover (one instruction, issued by wave 0), published to the
// other waves via s_wait_tensorcnt + workgroup barrier.
// ---------------------------------------------------------------------------
__global__ __launch_bounds__(GTHREADS) void gemm1(float* __restrict__ ws) {
  int b  = blockIdx.x / TILES;
  int nt = blockIdx.x % TILES;
  int n0 = nt * 16;
  int wave = threadIdx.x >> 5;      // m-tile index, 0..13
  int lane = threadIdx.x & 31;
  int m0 = wave * 16;

  const float* Df = ws + OFF_GT + (size_t)b * FF;
  __shared__ float sD[FDIM * 16];   // linear [k][c], written by TDM
  if (threadIdx.x < 32) {
    tdm_load_tile_f32(Df + n0, (unsigned)(uintptr_t)&sD[0], FDIM, 16, FDIM);
    __builtin_amdgcn_s_wait_tensorcnt(0);
  }
  __syncthreads();

  int ra = lane & 15, kk = (lane >> 4) * 2, cb = lane & 15;
  const float* Crow = ws + OFF_C + (size_t)(m0 + ra) * FDIM + kk;
  const float* Srow = ws + OFF_S + (size_t)(m0 + ra) * FDIM + kk;

  v8f accC = {}, accS = {};
  for (int k0 = 0; k0 < FDIM; k0 += 4) {
    if ((k0 & 63) == 0) {           // stream-ahead hint -> global_prefetch_b8
      __builtin_prefetch(Crow + k0 + 64, 0, 1);
      __builtin_prefetch(Srow + k0 + 64, 0, 1);
    }
    v2f ac, as2, bd;
    ac.x  = Crow[k0]; ac.y  = Crow[k0 + 1];
    as2.x = Srow[k0]; as2.y = Srow[k0 + 1];
    bd.x = sD[(k0 + kk + 0) * 16 + cb];
    bd.y = sD[(k0 + kk + 1) * 16 + cb];
    accC = __builtin_amdgcn_wmma_f32_16x16x4_f32(false, ac,  false, bd, (short)0, accC, false, false);
    accS = __builtin_amdgcn_wmma_f32_16x16x4_f32(false, as2, false, bd, (short)0, accS, false, false);
  }

  float* Ao = ws + OFF_A  + (size_t)b * FF;
  float* Bo = ws + OFF_BM + (size_t)b * FF;
  int rbase = m0 + ((lane >> 4) << 3);
  #pragma unroll
  for (int i = 0; i < 8; i++) {
    Ao[(rbase + i) * FDIM + n0 + cb] = accC[i];
    Bo[(rbase + i) * FDIM + n0 + cb] = accS[i];
  }
}

// ---------------------------------------------------------------------------
// Stage 2: Re = A*C - Bm*S, Im = A*S + Bm*C (C,S symmetric => F^T = F), with
// fused frequency weighting; C/S column tiles DMA'd to LDS by TDM. Per-wave
// partial sums written for a deterministic final reduction (no float atomics).
// ---------------------------------------------------------------------------
__global__ __launch_bounds__(GTHREADS) void gemm2(float* __restrict__ ws) {
  int b  = blockIdx.x / TILES;
  int nt = blockIdx.x % TILES;
  int n0 = nt * 16;
  int wave = threadIdx.x >> 5;
  int lane = threadIdx.x & 31;
  int m0 = wave * 16;

  const float* Cm = ws + OFF_C;
  const float* Sm = ws + OFF_S;
  __shared__ float sC[FDIM * 16];
  __shared__ float sS[FDIM * 16];
  if (threadIdx.x < 32) {
    tdm_load_tile_f32(Cm + n0, (unsigned)(uintptr_t)&sC[0], FDIM, 16, FDIM);
    tdm_load_tile_f32(Sm + n0, (unsigned)(uintptr_t)&sS[0], FDIM, 16, FDIM);
    __builtin_amdgcn_s_wait_tensorcnt(0);
  }
  __syncthreads();

  int ra = lane & 15, kk = (lane >> 4) * 2, cb = lane & 15;
  const float* Arow = ws + OFF_A  + (size_t)b * FF + (size_t)(m0 + ra) * FDIM + kk;
  const float* Brow = ws + OFF_BM + (size_t)b * FF + (size_t)(m0 + ra) * FDIM + kk;

  v8f acc_ac = {}, acc_bs = {}, acc_as = {}, acc_bc = {};
  for (int k0 = 0; k0 < FDIM; k0 += 4) {
    if ((k0 & 63) == 0) {
      __builtin_prefetch(Arow + k0 + 64, 0, 1);
      __builtin_prefetch(Brow + k0 + 64, 0, 1);
    }
    v2f aA, aB, bC, bS;
    aA.x = Arow[k0]; aA.y = Arow[k0 + 1];
    aB.x = Brow[k0]; aB.y = Brow[k0 + 1];
    bC.x = sC[(k0 + kk + 0) * 16 + cb];
    bC.y = sC[(k0 + kk + 1) * 16 + cb];
    bS.x = sS[(k0 + kk + 0) * 16 + cb];
    bS.y = sS[(k0 + kk + 1) * 16 + cb];
    acc_ac = __builtin_amdgcn_wmma_f32_16x16x4_f32(false, aA, false, bC, (short)0, acc_ac, false, false);
    acc_bs = __builtin_amdgcn_wmma_f32_16x16x4_f32(false, aB, false, bS, (short)0, acc_bs, false, false);
    acc_as = __builtin_amdgcn_wmma_f32_16x16x4_f32(false, aA, false, bS, (short)0, acc_as, false, false);
    acc_bc = __builtin_amdgcn_wmma_f32_16x16x4_f32(false, aB, false, bC, (short)0, acc_bc, false, false);
  }

  float local = 0.f;
  int rbase = m0 + ((lane >> 4) << 3);
  int kw = n0 + cb;
  float fw = (kw < FDIM / 2 ? (float)kw : (float)(kw - FDIM)) / (float)FDIM;
  #pragma unroll
  for (int i = 0; i < 8; i++) {
    int kh = rbase + i;
    float fh = (kh < FDIM / 2 ? (float)kh : (float)(kh - FDIM)) / (float)FDIM;
    float mag2 = fh * fh + fw * fw + EPS;
    float re = acc_ac[i] - acc_bs[i];
    float im = acc_as[i] + acc_bc[i];
    local += mag2 * (re * re + im * im);
  }
  // wave32 butterfly reduction, then one partial per wave (deterministic)
  for (int m = 16; m >= 1; m >>= 1) local += __shfl_xor(local, m, 32);
  if (lane == 0)
    ws[OFF_PART + ((size_t)b * TILES + nt) * TILES + wave] = local;
}

// Fixed-order reduction of the B*196 per-wave partials -> scalar loss.
__global__ void final_reduce(const float* __restrict__ ws, float* __restrict__ out) {
  __shared__ float red[256];
  int t = threadIdx.x;
  float s = 0.f;
  for (int i = t; i < B * TPB; i += 256) s += ws[OFF_PART + i];
  red[t] = s;
  __syncthreads();
  for (int w = 128; w >= 1; w >>= 1) {
    if (t < w) red[t] += red[t + w];
    __syncthreads();
  }
  if (t == 0)
    out[0] = red[0] * (1.0f / ((float)B * (float)FF * (float)FF)); // mean + Parseval (1/N) + (1/N)
}

extern "C" void kernel_launch(void* const* d_in, const int* in_sizes, int n_in,
                              void* d_out, int out_size, void* d_ws, size_t ws_size,
                              hipStream_t stream) {
  const float* pred = (const float*)d_in[0];
  const float* gt   = (const float*)d_in[1];
  const unsigned char* vmask = (const unsigned char*)d_in[2];
  float* ws  = (float*)d_ws;
  float* out = (float*)d_out;

  // zero the two field buffers (graph-capturable)
  (void)hipMemsetAsync(ws + OFF_GT, 0, (size_t)2 * B * FF * sizeof(float), stream);

  build_dft<<<(FF + 255) / 256, 256, 0, stream>>>(ws);
  rasterize<<<dim3(B * L, 2), 256, 0, stream>>>(pred, gt, vmask, ws);
  diff_kernel<<<(int)(((size_t)B * FF + 255) / 256), 256, 0, stream>>>(ws);
  gemm1<<<B * TILES, GTHREADS, 0, stream>>>(ws);
  gemm2<<<B * TILES, GTHREADS, 0, stream>>>(ws);
  final_reduce<<<1, 256, 0, stream>>>(ws, out);
}